// DGI_79628693668159
// MI455X (gfx1250) — compile-verified
//
#include <hip/hip_runtime.h>
#include <hip/hip_bf16.h>

#define N_NODES 100000
#define N_EDGES 1600000
#define D_IN 128
#define D_H 256
#define ROWS_PER_BLOCK 512

typedef __attribute__((ext_vector_type(2))) float v2f;
typedef __attribute__((ext_vector_type(8))) float v8f;

// ---------------------------------------------------------------------------
// Kernel 1: CSR row pointers from sorted adj_row via binary search (lower_bound)
// ---------------------------------------------------------------------------
__global__ void build_row_ptr(const int* __restrict__ adj_row, int* __restrict__ row_ptr) {
    int r = blockIdx.x * blockDim.x + threadIdx.x;
    if (r > N_NODES) return;
    int lo = 0, hi = N_EDGES;
    while (lo < hi) {
        int mid = (lo + hi) >> 1;
        if (adj_row[mid] < r) lo = mid + 1; else hi = mid;
    }
    row_ptr[r] = lo;
}

// ---------------------------------------------------------------------------
// Kernel 2: SpMM  agg[r,:] = sum_e vals[e] * feat[col[e],:]
// One 128-thread block per node; thread t owns feature dim t (coalesced).
// Deterministic (no atomics): edges of a row are summed in order by one block.
// ---------------------------------------------------------------------------
__global__ void spmm(const float* __restrict__ feat, const float* __restrict__ vals,
                     const int* __restrict__ col, const int* __restrict__ row_ptr,
                     float* __restrict__ agg) {
    const int r = blockIdx.x;
    const int t = threadIdx.x;            // 0..127 = feature dim
    const int e0 = row_ptr[r];
    const int e1 = row_ptr[r + 1];
    float acc = 0.f;
    for (int e = e0; e < e1; ++e) {
        const int c = col[e];
        const float w = vals[e];
        if (e + 1 < e1) {
            // CDNA5 global_prefetch_b8 of next gathered row
            __builtin_prefetch(&feat[(size_t)col[e + 1] * D_IN + t], 0, 1);
        }
        acc += w * feat[(size_t)c * D_IN + t];
    }
    agg[(size_t)r * D_IN + t] = acc;
}

// ---------------------------------------------------------------------------
// Kernel 3: h = PReLU(agg @ W1^T) via V_WMMA_F32_16X16X4_F32.
// One wave per 16x16 tile of h. B[k][n] = W1[n][k] (torch [out,in] layout), so
// A and B lane loads share the same index structure.
// Documented fp32 layouts (CDNA5 ISA 7.12.2):
//   A 16x4 : lanes 0-15 -> M=lane, v0=K0,v1=K1 ; lanes 16-31 -> v0=K2,v1=K3
//   C 16x16: vgpr i, lanes 0-15 -> M=i ; lanes 16-31 -> M=i+8 ; N=lane&15
// ---------------------------------------------------------------------------
__global__ void gemm_prelu_wmma(const float* __restrict__ agg, const float* __restrict__ W1,
                                const float* __restrict__ prelu_a, float* __restrict__ h) {
    const int wave = threadIdx.x >> 5;         // 4 waves / block
    const int lane = threadIdx.x & 31;
    const int m0 = blockIdx.x * 16;                       // row tile (6250 tiles exact)
    const int n0 = (blockIdx.y * 4 + wave) * 16;          // col tile (16 tiles = 256)
    const int ml   = lane & 15;
    const int koff = (lane >> 4) << 1;                    // 0 or 2

    const float* __restrict__ arow = agg + (size_t)(m0 + ml) * D_IN;
    const float* __restrict__ brow = W1  + (size_t)(n0 + ml) * D_IN;

    v8f c = {};
    #pragma unroll 4
    for (int k0 = 0; k0 < D_IN; k0 += 4) {
        v2f a, b;
        a.x = arow[k0 + koff];
        a.y = arow[k0 + koff + 1];
        b.x = brow[k0 + koff];
        b.y = brow[k0 + koff + 1];
        // D = A(16x4,f32) * B(4x16,f32) + C ; full fp32 precision
        c = __builtin_amdgcn_wmma_f32_16x16x4_f32(
                /*neg_a=*/false, a, /*neg_b=*/false, b,
                /*c_mod=*/(short)0, c, /*reuse_a=*/false, /*reuse_b=*/false);
    }

    const float slope = prelu_a[0];
    const int mbase = m0 + ((lane >> 4) << 3);            // +8 for upper half-lanes
    const int ncol  = n0 + ml;
    #pragma unroll
    for (int i = 0; i < 8; ++i) {
        float z = c[i];
        z = (z >= 0.f) ? z : slope * z;
        h[(size_t)(mbase + i) * D_H + ncol] = z;
    }
}

// ---------------------------------------------------------------------------
// Kernel 4: partial column sums of h (fixed-order, deterministic)
// ---------------------------------------------------------------------------
__global__ void colsum_partial(const float* __restrict__ h, float* __restrict__ partial) {
    const int t = threadIdx.x;                 // 0..255 = column
    const int r0 = blockIdx.x * ROWS_PER_BLOCK;
    int r1 = r0 + ROWS_PER_BLOCK; if (r1 > N_NODES) r1 = N_NODES;
    float s = 0.f;
    for (int r = r0; r < r1; ++r) s += h[(size_t)r * D_H + t];
    partial[(size_t)blockIdx.x * D_H + t] = s;
}

// ---------------------------------------------------------------------------
// Kernel 5: s = sigmoid(mean) ; v = w_bil @ s   (single block of 256 threads)
// ---------------------------------------------------------------------------
__global__ void finalize_sv(const float* __restrict__ partial, int nparts,
                            const float* __restrict__ w_bil, float* __restrict__ v) {
    __shared__ float s_sh[D_H];
    const int t = threadIdx.x;
    float s = 0.f;
    for (int p = 0; p < nparts; ++p) s += partial[(size_t)p * D_H + t];
    s *= (1.0f / (float)N_NODES);
    s_sh[t] = 1.0f / (1.0f + __expf(-s));
    __syncthreads();
    float acc = 0.f;
    #pragma unroll 8
    for (int j = 0; j < D_H; ++j) acc += w_bil[(size_t)t * D_H + j] * s_sh[j];
    v[t] = acc;
}

// ---------------------------------------------------------------------------
// Kernel 6: out[row] = dot(h[row,:], v)  — one wave32 per row, shfl_xor tree
// ---------------------------------------------------------------------------
__global__ void bilinear_dot(const float* __restrict__ h, const float* __restrict__ v,
                             float* __restrict__ out) {
    const int wave = threadIdx.x >> 5;
    const int lane = threadIdx.x & 31;
    const int row = blockIdx.x * (blockDim.x >> 5) + wave;
    if (row >= N_NODES) return;
    const float* __restrict__ hr = h + (size_t)row * D_H;
    float s = 0.f;
    #pragma unroll
    for (int j = 0; j < 8; ++j) s += hr[lane + 32 * j] * v[lane + 32 * j];
    #pragma unroll
    for (int off = 16; off > 0; off >>= 1) s += __shfl_xor(s, off, 32);
    if (lane == 0) out[row] = s;
}

// ---------------------------------------------------------------------------
extern "C" void kernel_launch(void* const* d_in, const int* in_sizes, int n_in,
                              void* d_out, int out_size, void* d_ws, size_t ws_size,
                              hipStream_t stream) {
    const float* x        = (const float*)d_in[0];
    const float* x_tilde  = (const float*)d_in[1];
    const float* adj_vals = (const float*)d_in[2];
    const int*   adj_row  = (const int*)  d_in[3];
    const int*   adj_col  = (const int*)  d_in[4];
    const float* W1       = (const float*)d_in[5];
    const float* prelu_a  = (const float*)d_in[6];
    const float* w_bil    = (const float*)d_in[7];
    float* out = (float*)d_out;

    // workspace carving (peak ~154 MB: row_ptr + agg + h + partials + v)
    char* ws = (char*)d_ws;
    size_t off = 0;
    int* row_ptr = (int*)(ws + off);
    off += (((size_t)(N_NODES + 1) * sizeof(int)) + 4095) & ~(size_t)4095;
    float* agg = (float*)(ws + off);  off += (size_t)N_NODES * D_IN * sizeof(float);
    float* h   = (float*)(ws + off);  off += (size_t)N_NODES * D_H  * sizeof(float);
    const int nparts = (N_NODES + ROWS_PER_BLOCK - 1) / ROWS_PER_BLOCK;
    float* partial = (float*)(ws + off); off += (size_t)nparts * D_H * sizeof(float);
    float* v = (float*)(ws + off);       off += (size_t)D_H * sizeof(float);

    build_row_ptr<<<(N_NODES + 256) / 256, 256, 0, stream>>>(adj_row, row_ptr);

    const dim3 ggrid(N_NODES / 16, 4);   // 6250 x 4 blocks, 4 waves each

    // ---- positive branch: h, readout, dp ----
    spmm<<<N_NODES, D_IN, 0, stream>>>(x, adj_vals, adj_col, row_ptr, agg);
    gemm_prelu_wmma<<<ggrid, 128, 0, stream>>>(agg, W1, prelu_a, h);
    colsum_partial<<<nparts, D_H, 0, stream>>>(h, partial);
    finalize_sv<<<1, D_H, 0, stream>>>(partial, nparts, w_bil, v);
    bilinear_dot<<<N_NODES / 8, 256, 0, stream>>>(h, v, out);

    // ---- negative branch (reuse agg/h buffers): dn ----
    spmm<<<N_NODES, D_IN, 0, stream>>>(x_tilde, adj_vals, adj_col, row_ptr, agg);
    gemm_prelu_wmma<<<ggrid, 128, 0, stream>>>(agg, W1, prelu_a, h);
    bilinear_dot<<<N_NODES / 8, 256, 0, stream>>>(h, v, out + N_NODES);
}